// Pairer_48447231099129
// MI455X (gfx1250) — compile-verified
//
#include <hip/hip_runtime.h>

#define B_ 8
#define N_ 128
#define H_ 256
#define PITCH 260   // ≡ 4 (mod 64 banks): 16B-aligned rows (async B128 OK) and
                    // conflict-free WMMA fragment reads (4m+kb vs 4m+kb+2, disjoint mod 4)

typedef __attribute__((ext_vector_type(2))) float v2f;
typedef __attribute__((ext_vector_type(8))) float v8f;

// ---------------------------------------------------------------------------
// Kernel 1: per-row dot products  rdot[b,i] = x[b,i]·W1 , cdot[b,i] = x[b,i]·W2
// One wave32 per row; 8 waves per 256-thread block.
// ---------------------------------------------------------------------------
__global__ void __launch_bounds__(256)
row_dots_kernel(const float* __restrict__ x,
                const float* __restrict__ W,
                float* __restrict__ rdot,
                float* __restrict__ cdot) {
    int wave = threadIdx.x >> 5;
    int lane = threadIdx.x & 31;
    int row  = blockIdx.x * 8 + wave;          // 0 .. B*N-1 (exact)
    const float* xr = x + (size_t)row * H_;
    float d1 = 0.f, d2 = 0.f;
    for (int h = lane; h < H_; h += 32) {
        float xv = xr[h];
        d1 += xv * W[h];          // W1
        d2 += xv * W[H_ + h];     // W2
    }
    for (int off = 16; off; off >>= 1) {
        d1 += __shfl_xor(d1, off, 32);
        d2 += __shfl_xor(d2, off, 32);
    }
    if (lane == 0) { rdot[row] = d1; cdot[row] = d2; }
}

// ---------------------------------------------------------------------------
// Kernel 2: one wave32 per 16x16 output tile.
//   quad = (x[b] ⊙ W3) @ x[b]^T via V_WMMA_F32_16X16X4_F32, K-loop of 64.
//   B slab staged via GLOBAL_LOAD_ASYNC_TO_LDS_B128 (ASYNCcnt), overlapped with
//   the VGPR-path A slab staging (needs the ×W3 multiply).
//   Epilogue fuses rank-1 terms, rel-pos W4 (torch-faithful index), bias, mask.
// ---------------------------------------------------------------------------
__global__ void __launch_bounds__(32)
pair_tile_kernel(const float* __restrict__ x,
                 const unsigned char* __restrict__ mask,   // jnp bool = 1 byte
                 const float* __restrict__ W,
                 const float* __restrict__ bias,
                 const float* __restrict__ rdot,
                 const float* __restrict__ cdot,
                 float* __restrict__ logits) {
    __shared__ float aS[16][PITCH];
    __shared__ float bS[16][PITCH];

    int tile = blockIdx.x;          // B * 8 * 8 = 512 tiles
    int b  = tile >> 6;
    int it = (tile >> 3) & 7;
    int jt = tile & 7;
    int i0 = it * 16, j0 = jt * 16;
    int lane = threadIdx.x;

    const float* W3 = W + 2 * H_;
    const float* xb = x + (size_t)b * N_ * H_;

    // Phase 1: issue async global->LDS copies for the raw B slab.
    // 16 rows x 64 float4 chunks = 1024 chunks; 32 per lane, 16B each.
    for (int t = 0; t < 32; ++t) {
        int idx = lane + t * 32;
        int r   = idx >> 6;                    // chunk row
        int kc  = (idx & 63) * 4;              // chunk column (floats)
        unsigned lds_b =
            (unsigned)(unsigned long long)(uintptr_t)&bS[r][kc];   // LDS byte addr
        unsigned long long ga =
            (unsigned long long)(uintptr_t)(xb + (size_t)(j0 + r) * H_ + kc);
        asm volatile("global_load_async_to_lds_b128 %0, %1, off"
                     :: "v"(lds_b), "v"(ga) : "memory");
    }

    // Phase 2 (overlaps the async engine): stage the W3-scaled A slab via VGPRs.
    for (int idx = lane; idx < 16 * (H_ / 4); idx += 32) {
        int r  = idx >> 6;
        int kc = (idx & 63) * 4;
        float4 av = *(const float4*)(xb + (size_t)(i0 + r) * H_ + kc);
        float4 wv = *(const float4*)(W3 + kc);
        aS[r][kc + 0] = av.x * wv.x;
        aS[r][kc + 1] = av.y * wv.y;
        aS[r][kc + 2] = av.z * wv.z;
        aS[r][kc + 3] = av.w * wv.w;
    }

    // Drain the async copies, then normal LDS visibility.
    asm volatile("s_wait_asynccnt 0x0" ::: "memory");
    __syncthreads();

    // ISA A 16x4 layout: lanes 0-15 -> M=lane, K=k..k+1 ; lanes 16-31 -> K=k+2..k+3.
    // B 4x16 mirrored: lane -> N=lane&15, same K split.
    int half = lane >> 4;
    int m    = lane & 15;
    v8f acc = {};
    for (int k = 0; k < H_; k += 4) {
        int kb = k + half * 2;
        v2f af, bf;
        af.x = aS[m][kb];  af.y = aS[m][kb + 1];
        bf.x = bS[m][kb];  bf.y = bS[m][kb + 1];
        acc = __builtin_amdgcn_wmma_f32_16x16x4_f32(
            /*neg_a=*/false, af, /*neg_b=*/false, bf,
            /*c_mod=*/(short)0, acc, /*reuse_a=*/false, /*reuse_b=*/false);
    }

    // Epilogue. C/D layout: VGPR v -> M = v + 8*half, N = lane&15.
    float bb  = bias[0];
    int   col = j0 + m;
    bool  mj  = mask[b * N_ + col] != 0;
    float cj  = cdot[b * N_ + col];
    const float* W4 = W + 3 * H_;            // length 2N-1 = 255
    for (int v = 0; v < 8; ++v) {
        int   row  = i0 + v + half * 8;
        // torch-faithful repeat_interleave(dim=0).view quirk:
        // one_hots[b,i] = oh[(b*N+i)//B] -> s_eff = b*16 + i/8
        int   seff = b * 16 + (row >> 3);
        float val  = acc[v]
                   + rdot[b * N_ + row]
                   + cj
                   + W4[col - seff + (N_ - 1)]
                   + bb;
        bool  mi = mask[b * N_ + row] != 0;
        logits[((size_t)b * N_ + row) * N_ + col] =
            (mi && mj) ? val : -__builtin_inff();
    }
}

// ---------------------------------------------------------------------------
// Kernel 3: preds[b,i] = argmax_j logits[b,i,j], first-occurrence tie-break
// (matches jnp.argmax; all -inf rows -> 0). One wave32 per row.
// ---------------------------------------------------------------------------
__global__ void __launch_bounds__(256)
argmax_kernel(const float* __restrict__ logits, int* __restrict__ preds) {
    int wave = threadIdx.x >> 5;
    int lane = threadIdx.x & 31;
    int row  = blockIdx.x * 8 + wave;          // 0 .. B*N-1
    const float* lr = logits + (size_t)row * N_;
    float bv = -__builtin_inff();
    int   bi = 0;
    for (int j = lane; j < N_; j += 32) {
        float v = lr[j];
        if (v > bv || (v == bv && j < bi)) { bv = v; bi = j; }
    }
    for (int off = 16; off; off >>= 1) {
        float ov = __shfl_xor(bv, off, 32);
        int   oi = __shfl_xor(bi, off, 32);
        if (ov > bv || (ov == bv && oi < bi)) { bv = ov; bi = oi; }
    }
    if (lane == 0) preds[row] = bi;
}

// ---------------------------------------------------------------------------
extern "C" void kernel_launch(void* const* d_in, const int* in_sizes, int n_in,
                              void* d_out, int out_size, void* d_ws, size_t ws_size,
                              hipStream_t stream) {
    const float*         x    = (const float*)d_in[0];          // [B,N,H]
    const unsigned char* mask = (const unsigned char*)d_in[1];  // [B,N] bool
    const float*         W    = (const float*)d_in[2];          // [1,1023]
    const float*         bias = (const float*)d_in[3];          // [1]

    float* logits = (float*)d_out;                        // B*N*N f32
    int*   preds  = (int*)d_out + (size_t)B_ * N_ * N_;   // B*N i32

    float* rdot = (float*)d_ws;                           // B*N
    float* cdot = rdot + B_ * N_;                         // B*N

    row_dots_kernel<<<B_ * N_ / 8, 256, 0, stream>>>(x, W, rdot, cdot);
    pair_tile_kernel<<<B_ * 8 * 8, 32, 0, stream>>>(x, mask, W, bias,
                                                    rdot, cdot, logits);
    argmax_kernel<<<B_ * N_ / 8, 256, 0, stream>>>(logits, preds);
}